// TTTWrapper_47588237640299
// MI455X (gfx1250) — compile-verified
//
#include <hip/hip_runtime.h>
#include <math.h>

typedef float v2f __attribute__((ext_vector_type(2)));
typedef float v8f __attribute__((ext_vector_type(8)));

#define B_      2
#define T_      4096
#define DM      1024
#define DI      2048
#define CHUNK_  256
#define NCHUNK  (T_ / CHUNK_)
#define KSIZE_  5
#define EPS_    1e-6f
#define DW_CAP  0.1f
#define G_CAP   0.02f

// ---- workspace layout (float offsets) ----
#define WS_W0SQ   0
#define WS_W0N    1
#define WS_ETA    2
#define WS_DECAY  3
#define WS_NORMG  4   // [B]
#define WS_NORMP  6   // [B]
#define WS_S      8   // [B]
#define WS_P      16                 // unscaled deltaW state, B*DM*DI
#define P_ELEMS   (B_ * DM * DI)     // 4,194,304
#define WS_G      (WS_P + P_ELEMS)   // raw G scratch, B*DM*DI
#define WS_VU     (WS_G + P_ELEMS)   // vu (conv + double rmsnorm), B*T*DM
#define VU_ELEMS  (B_ * T_ * DM)
#define WS_ZRMS   (WS_VU + VU_ELEMS) // per-row rsqrt(mean(z^2)+eps), B*T
// total ~16.79M floats ~= 67.2 MB

// ---------------- init / reductions ----------------

__global__ void k_zero(float* __restrict__ ws) {
    size_t n = (size_t)WS_P + P_ELEMS;
    for (size_t i = blockIdx.x * (size_t)blockDim.x + threadIdx.x; i < n;
         i += (size_t)gridDim.x * blockDim.x)
        ws[i] = 0.0f;
}

__global__ void k_w0sq(const float* __restrict__ W0, float* __restrict__ ws) {
    const float4* W4 = (const float4*)W0;
    size_t n4 = (size_t)DM * DI / 4;
    float acc = 0.0f;
    for (size_t i = blockIdx.x * (size_t)blockDim.x + threadIdx.x; i < n4;
         i += (size_t)gridDim.x * blockDim.x) {
        float4 v = W4[i];
        acc += v.x * v.x + v.y * v.y + v.z * v.z + v.w * v.w;
    }
    for (int off = 16; off > 0; off >>= 1) acc += __shfl_xor(acc, off, 32);
    if ((threadIdx.x & 31) == 0) atomicAdd(&ws[WS_W0SQ], acc);
}

__global__ void k_params(const float* __restrict__ loglr,
                         const float* __restrict__ logdec,
                         float* __restrict__ ws) {
    float eta = expf(loglr[0]);
    float sg  = 1.0f / (1.0f + expf(-logdec[0]));
    ws[WS_ETA]   = eta;
    ws[WS_DECAY] = 0.9f + (0.995f - 0.9f) * sg;
    ws[WS_W0N]   = sqrtf(ws[WS_W0SQ]);
}

// causal depthwise conv (K=5) + rmsnorm applied twice -> vu.  One block per (b,t).
__global__ void k_conv_vu(const float* __restrict__ src,  // (B,T,DM)
                          const float* __restrict__ w,    // (DM,5)
                          float* __restrict__ ws) {
    __shared__ float red[256];
    int blk = blockIdx.x, b = blk / T_, t = blk % T_;
    const float* sb = src + ((size_t)b * T_) * DM;
    float y[4];
    float ssq = 0.0f;
    #pragma unroll
    for (int i = 0; i < 4; i++) {
        int c = threadIdx.x + 256 * i;
        float acc = 0.0f;
        #pragma unroll
        for (int j = 0; j < KSIZE_; j++) {
            int tj = t + j - (KSIZE_ - 1);
            if (tj >= 0) acc += sb[(size_t)tj * DM + c] * w[c * KSIZE_ + j];
        }
        y[i] = acc;
        ssq += acc * acc;
    }
    red[threadIdx.x] = ssq;
    __syncthreads();
    for (int s = 128; s > 0; s >>= 1) {
        if (threadIdx.x < s) red[threadIdx.x] += red[threadIdx.x + s];
        __syncthreads();
    }
    float ms  = red[0] / (float)DM;
    float rs1 = rsqrtf(ms + EPS_);
    float ms2 = ms / (ms + EPS_);           // mean-square after first norm
    float rs2 = rsqrtf(ms2 + EPS_);         // second rmsnorm (vu = rmsnorm(vhat))
    float* vu = ws + WS_VU + ((size_t)b * T_ + t) * DM;
    #pragma unroll
    for (int i = 0; i < 4; i++) vu[threadIdx.x + 256 * i] = y[i] * rs1 * rs2;
}

// per-row rmsnorm scalar for z. One block per (b,t).
__global__ void k_zrms(const float* __restrict__ z, float* __restrict__ ws) {
    __shared__ float red[256];
    int blk = blockIdx.x, b = blk / T_, t = blk % T_;
    const float* zr = z + ((size_t)b * T_ + t) * DI;
    float ssq = 0.0f;
    #pragma unroll
    for (int i = 0; i < 8; i++) {
        float v = zr[threadIdx.x + 256 * i];
        ssq += v * v;
    }
    red[threadIdx.x] = ssq;
    __syncthreads();
    for (int s = 128; s > 0; s >>= 1) {
        if (threadIdx.x < s) red[threadIdx.x] += red[threadIdx.x + s];
        __syncthreads();
    }
    if (threadIdx.x == 0)
        ws[WS_ZRMS + (size_t)b * T_ + t] = rsqrtf(red[0] / (float)DI + EPS_);
}

// s[b] = min(DW_CAP * ||W0|| / (||P|| + 1e-8), 1); reset norm accumulators.
__global__ void k_scale(float* __restrict__ ws) {
    int i = threadIdx.x;
    if (i < B_) {
        float np = ws[WS_NORMP + i];
        ws[WS_S + i] = fminf(DW_CAP * ws[WS_W0N] / (sqrtf(np) + 1e-8f), 1.0f);
        ws[WS_NORMP + i] = 0.0f;
        ws[WS_NORMG + i] = 0.0f;
    }
}

// ---------------- GEMM 1: o = z_c (W0 + s*P)^T + bias, WMMA f32 16x16x4 ----------------
// wave tile: 16 (t) x 64 (m).  512 waves total; blockDim=128 (4 waves), grid=128.
__global__ void k_gemm_o(const float* __restrict__ z,
                         const float* __restrict__ W0,
                         const float* __restrict__ bias,
                         const float* __restrict__ ws,
                         float* __restrict__ out, int chunk) {
    int wave = (blockIdx.x * blockDim.x + threadIdx.x) >> 5;
    int lane = threadIdx.x & 31;
    int b    = wave >> 8;          // 256 waves per batch
    int rem  = wave & 255;
    int tt   = rem >> 4, mt = rem & 15;
    int t0   = chunk * CHUNK_ + tt * 16;
    int m0   = mt * 64;

    float s = ws[WS_S + b];
    const float* P  = ws + WS_P + (size_t)b * DM * DI;
    const float* zb = z + ((size_t)b * T_ + t0) * DI;

    int row = lane & 15;           // A: M index (t); B: N index (m)
    int kb  = (lane >> 4) * 2;     // K sub-offset per half-wave

    v8f c0 = {}, c1 = {}, c2 = {}, c3 = {};
    for (int k = 0; k < DI; k += 4) {
        v2f a;
        a.x = zb[(size_t)row * DI + k + kb];
        a.y = zb[(size_t)row * DI + k + kb + 1];
        #pragma unroll
        for (int j = 0; j < 4; j++) {
            int m = m0 + j * 16 + row;
            const float* wr = W0 + (size_t)m * DI + k + kb;
            const float* pr = P  + (size_t)m * DI + k + kb;
            v2f bf;
            bf.x = wr[0] + s * pr[0];
            bf.y = wr[1] + s * pr[1];
            switch (j) {
            case 0: c0 = __builtin_amdgcn_wmma_f32_16x16x4_f32(false, a, false, bf, (short)0, c0, false, false); break;
            case 1: c1 = __builtin_amdgcn_wmma_f32_16x16x4_f32(false, a, false, bf, (short)0, c1, false, false); break;
            case 2: c2 = __builtin_amdgcn_wmma_f32_16x16x4_f32(false, a, false, bf, (short)0, c2, false, false); break;
            case 3: c3 = __builtin_amdgcn_wmma_f32_16x16x4_f32(false, a, false, bf, (short)0, c3, false, false); break;
            }
        }
    }
    int hi = (lane >> 4) * 8;
    #pragma unroll
    for (int j = 0; j < 4; j++) {
        v8f c = (j == 0) ? c0 : (j == 1) ? c1 : (j == 2) ? c2 : c3;
        int m = m0 + j * 16 + row;
        float bv = bias[m];
        #pragma unroll
        for (int r = 0; r < 8; r++) {
            int t = t0 + r + hi;
            out[((size_t)b * T_ + t) * DM + m] = c[r] + bv;
        }
    }
}

// ---------------- GEMM 2: G = vu^T zu / C, + ||G||^2 accumulation ----------------
// wave tile: 16 (m) x 64 (d). 4096 waves; blockDim=128, grid=1024.
__global__ void k_gemm_g(const float* __restrict__ z,
                         float* __restrict__ ws, int chunk) {
    int wave = (blockIdx.x * blockDim.x + threadIdx.x) >> 5;
    int lane = threadIdx.x & 31;
    int b    = wave >> 11;         // 2048 waves per batch
    int rem  = wave & 2047;
    int mt   = rem >> 5, dt = rem & 31;
    int m0   = mt * 16;
    int d0   = dt * 64;
    int t0   = chunk * CHUNK_;

    const float* vu = ws + WS_VU + (size_t)b * T_ * DM;
    const float* zr = ws + WS_ZRMS + (size_t)b * T_;
    const float* zb = z + (size_t)b * T_ * DI;

    int col = lane & 15;           // A: M (m); B: N (d)
    int kb  = (lane >> 4) * 2;

    v8f c0 = {}, c1 = {}, c2 = {}, c3 = {};
    for (int k = 0; k < CHUNK_; k += 4) {
        int tA = t0 + k + kb;
        v2f a;                      // A[m][t] = vu[t][m]  (lane-contiguous in m)
        a.x = vu[(size_t)tA * DM + m0 + col];
        a.y = vu[(size_t)(tA + 1) * DM + m0 + col];
        float r0 = zr[tA], r1 = zr[tA + 1];
        #pragma unroll
        for (int j = 0; j < 4; j++) {
            v2f bf;                 // B[t][d] = z[t][d] * zrms[t] (lane-contiguous in d)
            bf.x = zb[(size_t)tA * DI + d0 + j * 16 + col] * r0;
            bf.y = zb[(size_t)(tA + 1) * DI + d0 + j * 16 + col] * r1;
            switch (j) {
            case 0: c0 = __builtin_amdgcn_wmma_f32_16x16x4_f32(false, a, false, bf, (short)0, c0, false, false); break;
            case 1: c1 = __builtin_amdgcn_wmma_f32_16x16x4_f32(false, a, false, bf, (short)0, c1, false, false); break;
            case 2: c2 = __builtin_amdgcn_wmma_f32_16x16x4_f32(false, a, false, bf, (short)0, c2, false, false); break;
            case 3: c3 = __builtin_amdgcn_wmma_f32_16x16x4_f32(false, a, false, bf, (short)0, c3, false, false); break;
            }
        }
    }
    float* G = ws + WS_G + (size_t)b * DM * DI;
    const float invC = 1.0f / (float)CHUNK_;
    int hi = (lane >> 4) * 8;
    float ssq = 0.0f;
    #pragma unroll
    for (int j = 0; j < 4; j++) {
        v8f c = (j == 0) ? c0 : (j == 1) ? c1 : (j == 2) ? c2 : c3;
        #pragma unroll
        for (int r = 0; r < 8; r++) {
            float g = c[r] * invC;
            int m = m0 + r + hi;
            int d = d0 + j * 16 + col;
            G[(size_t)m * DI + d] = g;
            ssq += g * g;
        }
    }
    for (int off = 16; off > 0; off >>= 1) ssq += __shfl_xor(ssq, off, 32);
    if (lane == 0) atomicAdd(&ws[WS_NORMG + b], ssq);
}

// ---------------- state update: P <- decay*s*P + (1-decay)*eta*g*G ; accumulate ||P||^2 ----
__global__ void k_update(float* __restrict__ ws) {
    int b = blockIdx.y;
    float W0n   = ws[WS_W0N];
    float eta   = ws[WS_ETA];
    float decay = ws[WS_DECAY];
    float s     = ws[WS_S + b];
    float ng    = ws[WS_NORMG + b];
    float g     = fminf(G_CAP * W0n / (sqrtf(ng) + 1e-8f), 1.0f);
    float ca = decay * s;
    float cg = (1.0f - decay) * eta * g;

    float4* P4 = (float4*)(ws + WS_P + (size_t)b * DM * DI);
    const float4* G4 = (const float4*)(ws + WS_G + (size_t)b * DM * DI);
    size_t n4 = (size_t)DM * DI / 4;
    float ssq = 0.0f;
    for (size_t i = blockIdx.x * (size_t)blockDim.x + threadIdx.x; i < n4;
         i += (size_t)gridDim.x * blockDim.x) {
        float4 p = P4[i];
        float4 q = G4[i];
        float4 pn;
        pn.x = ca * p.x + cg * q.x;
        pn.y = ca * p.y + cg * q.y;
        pn.z = ca * p.z + cg * q.z;
        pn.w = ca * p.w + cg * q.w;
        P4[i] = pn;
        ssq += pn.x * pn.x + pn.y * pn.y + pn.z * pn.z + pn.w * pn.w;
    }
    for (int off = 16; off > 0; off >>= 1) ssq += __shfl_xor(ssq, off, 32);
    if ((threadIdx.x & 31) == 0) atomicAdd(&ws[WS_NORMP + b], ssq);
}

// ---------------- finalize: deltaW_out = s[b] * P ----------------
__global__ void k_final(const float* __restrict__ ws, float* __restrict__ dW) {
    size_t n = P_ELEMS;
    for (size_t i = blockIdx.x * (size_t)blockDim.x + threadIdx.x; i < n;
         i += (size_t)gridDim.x * blockDim.x) {
        int b = (int)(i / ((size_t)DM * DI));
        dW[i] = ws[WS_S + b] * ws[WS_P + i];
    }
}

extern "C" void kernel_launch(void* const* d_in, const int* in_sizes, int n_in,
                              void* d_out, int out_size, void* d_ws, size_t ws_size,
                              hipStream_t stream) {
    (void)in_sizes; (void)n_in; (void)out_size; (void)ws_size;
    const float* z     = (const float*)d_in[0];
    const float* src   = (const float*)d_in[1];
    const float* W0    = (const float*)d_in[2];
    const float* bias  = (const float*)d_in[3];
    const float* convw = (const float*)d_in[4];
    const float* loglr = (const float*)d_in[5];
    const float* logdc = (const float*)d_in[6];
    float* out = (float*)d_out;                         // (B,T,DM)
    float* dW  = out + (size_t)B_ * T_ * DM;            // (B,DM,DI)
    float* ws  = (float*)d_ws;

    k_zero  <<<4096, 256, 0, stream>>>(ws);
    k_w0sq  <<<1024, 256, 0, stream>>>(W0, ws);
    k_params<<<1, 1,    0, stream>>>(loglr, logdc, ws);
    k_conv_vu<<<B_ * T_, 256, 0, stream>>>(src, convw, ws);
    k_zrms  <<<B_ * T_, 256, 0, stream>>>(z, ws);

    for (int chunk = 0; chunk < NCHUNK; chunk++) {
        k_scale <<<1, 32, 0, stream>>>(ws);
        k_gemm_o<<<128, 128, 0, stream>>>(z, W0, bias, ws, out, chunk);
        k_gemm_g<<<1024, 128, 0, stream>>>(z, ws, chunk);
        k_update<<<dim3(2048, B_), 256, 0, stream>>>(ws);
    }
    k_scale<<<1, 32, 0, stream>>>(ws);
    k_final<<<4096, 256, 0, stream>>>(ws, dW);
}